// NegativeSoftmax_4303557231312
// MI455X (gfx1250) — compile-verified
//
#include <hip/hip_runtime.h>
#include <hip/hip_bf16.h>

typedef __attribute__((ext_vector_type(2))) float v2f;
typedef __attribute__((ext_vector_type(8))) float v8f;

#define CCH   2048
#define NSUP  25
#define NQRY  600
#define NWAY  5

// ---------------------------------------------------------------------------
// Kernel 1: spatial mean-pool  (s,c,5,5) -> (s,c).  Pure bandwidth (~123MB
// read @ 23.3 TB/s ~= 5us floor; each wave covers 3.2KB contiguous).
// ---------------------------------------------------------------------------
__global__ __launch_bounds__(256) void pool_kernel(
    const float* __restrict__ sup, const float* __restrict__ qry,
    float* __restrict__ sx, float* __restrict__ qx, int total)
{
    int o = blockIdx.x * 256 + threadIdx.x;
    if (o >= total) return;
    int s = o >> 11;          // / 2048
    int c = o & (CCH - 1);    // % 2048
    const float* src;
    float* dst;
    if (s < NSUP) {
        src = sup + (size_t)(s * CCH + c) * 25;
        dst = sx + s * CCH + c;
    } else {
        int q = s - NSUP;
        src = qry + (size_t)(q * CCH + c) * 25;
        dst = qx + q * CCH + c;
    }
    float acc = 0.f;
#pragma unroll
    for (int u = 0; u < 25; ++u) acc += src[u];
    *dst = acc * (1.0f / 25.0f);
}

// ---------------------------------------------------------------------------
// Kernel 2: 700 sequential SGD+momentum steps. Single workgroup (256 thr,
// 8 waves). W and momentum live in registers: thread t owns columns
// k = t + 256*j (j=0..7) for all 5 classes. Forward z = x@W^T reduced via
// wave32 shfl_xor then an 8-wave LDS combine. Softmax grad:
//   coef[i][c] = SCALE * (m_i/Msum) * (p[i][c] - onehot);  (z.min term has
//   zero gradient and softmax is shift-invariant, so it is dropped.)
// ---------------------------------------------------------------------------
__global__ __launch_bounds__(256) void train_kernel(
    const float* __restrict__ sx, const int* __restrict__ sy,
    const int* __restrict__ bids, const int* __restrict__ bmask,
    const float* __restrict__ w0, float* __restrict__ wout, int n_steps)
{
    const int t = threadIdx.x;
    const int lane = t & 31;
    const int wave = t >> 5;

    __shared__ float zpart[8 * 20];
    __shared__ float coef[20];

    float Wr[NWAY][8], Br[NWAY][8];
#pragma unroll
    for (int c = 0; c < NWAY; ++c)
#pragma unroll
        for (int j = 0; j < 8; ++j) {
            Wr[c][j] = w0[c * CCH + t + 256 * j];
            Br[c][j] = 0.f;
        }

    for (int s = 0; s < n_steps; ++s) {
        int ids[4], yv[4];
        float mk[4];
#pragma unroll
        for (int i = 0; i < 4; ++i) {
            ids[i] = bids[s * 4 + i];
            mk[i]  = bmask[s * 4 + i] ? 1.f : 0.f;
            yv[i]  = sy[ids[i]];
        }
        float xr[4][8];
#pragma unroll
        for (int i = 0; i < 4; ++i)
#pragma unroll
            for (int j = 0; j < 8; ++j)
                xr[i][j] = sx[ids[i] * CCH + t + 256 * j];

        // partial z[i][c] over this thread's 8 columns, then wave reduce
        float zp[4][NWAY];
#pragma unroll
        for (int i = 0; i < 4; ++i)
#pragma unroll
            for (int c = 0; c < NWAY; ++c) {
                float acc = 0.f;
#pragma unroll
                for (int j = 0; j < 8; ++j) acc = fmaf(xr[i][j], Wr[c][j], acc);
#pragma unroll
                for (int o = 16; o > 0; o >>= 1) acc += __shfl_xor(acc, o, 32);
                zp[i][c] = acc;
            }
        if (lane == 0) {
#pragma unroll
            for (int i = 0; i < 4; ++i)
#pragma unroll
                for (int c = 0; c < NWAY; ++c)
                    zpart[wave * 20 + i * 5 + c] = zp[i][c];
        }
        __syncthreads();

        if (t < 4) {
            const int i = t;
            float z[NWAY];
#pragma unroll
            for (int c = 0; c < NWAY; ++c) {
                float acc = 0.f;
#pragma unroll
                for (int w = 0; w < 8; ++w) acc += zpart[w * 20 + i * 5 + c];
                z[c] = acc;
            }
            float msum = mk[0] + mk[1] + mk[2] + mk[3];
            float l[NWAY], mx = -1e30f;
#pragma unroll
            for (int c = 0; c < NWAY; ++c) {
                l[c] = (z[c] - ((c == yv[i]) ? 0.4f : 0.f)) * 10.0f;
                mx = fmaxf(mx, l[c]);
            }
            float es = 0.f;
#pragma unroll
            for (int c = 0; c < NWAY; ++c) { l[c] = __expf(l[c] - mx); es += l[c]; }
            float sc = 10.0f * (mk[i] / msum);
#pragma unroll
            for (int c = 0; c < NWAY; ++c)
                coef[i * 5 + c] = sc * (l[c] / es - ((c == yv[i]) ? 1.f : 0.f));
        }
        __syncthreads();

        float cf[4][NWAY];
#pragma unroll
        for (int i = 0; i < 4; ++i)
#pragma unroll
            for (int c = 0; c < NWAY; ++c) cf[i][c] = coef[i * 5 + c];

#pragma unroll
        for (int c = 0; c < NWAY; ++c)
#pragma unroll
            for (int j = 0; j < 8; ++j) {
                float g = 0.f;
#pragma unroll
                for (int i = 0; i < 4; ++i) g = fmaf(cf[i][c], xr[i][j], g);
                float d = g + 0.001f * Wr[c][j];              // + WD*W
                Br[c][j] = (s == 0) ? d : fmaf(0.9f, Br[c][j], 0.1f * d);
                Wr[c][j] -= 1.01f * Br[c][j];                 // LR
            }
        __syncthreads();   // protect zpart/coef for next step
    }

#pragma unroll
    for (int c = 0; c < NWAY; ++c)
#pragma unroll
        for (int j = 0; j < 8; ++j)
            wout[c * CCH + t + 256 * j] = Wr[c][j];
}

// ---------------------------------------------------------------------------
// Kernel 3: query scoring via V_WMMA_F32_16X16X4_F32.
// One wave per 16-query tile. A = 16x4 chunk of qx; B = 4x16 chunk of W^T.
// No operand masking needed: a clamped A-row only pollutes its own D-row
// (rows >= 600 are never stored) and a clamped B-column only pollutes
// D-columns 5..15 (never read by the argmax) -- all clamped sources are
// finite real data, so the K-loop is pure loads + WMMA.
// D spilled via LDS for the per-row argmax (min-shift and *SCALE are
// monotone -> argmax unchanged).
// ---------------------------------------------------------------------------
__global__ __launch_bounds__(32) void score_kernel(
    const float* __restrict__ qx, const float* __restrict__ W,
    const int* __restrict__ qy, float* __restrict__ out, int Q)
{
    __shared__ float Wlds[NWAY * CCH];
    __shared__ float tile[16 * 16];

    const int lane = threadIdx.x;
    for (int i = lane; i < NWAY * CCH; i += 32) Wlds[i] = W[i];
    __syncthreads();

    const int tidx = blockIdx.x;
    const int half = lane >> 4;          // 0: K0/K1, 1: K2/K3
    const int n    = lane & 15;          // row-in-tile (A) / class (B)
    const int row  = tidx * 16 + n;
    const float* arow = qx + (size_t)((row < Q) ? row : 0) * CCH;
    const float* brow = Wlds + ((n < NWAY) ? n : 0) * CCH;
    const int kph = half * 2;

    v8f acc = {};
#pragma unroll 4
    for (int k = 0; k < CCH; k += 4) {
        v2f a = *(const v2f*)(arow + k + kph);
        v2f b = *(const v2f*)(brow + k + kph);
        // 8 args: (neg_a, A, neg_b, B, c_mod, C, reuse_a, reuse_b)
        acc = __builtin_amdgcn_wmma_f32_16x16x4_f32(
            false, a, false, b, (short)0, acc, false, false);
    }

    // D layout: VGPR r, lanes 0-15 -> M=r, N=lane; lanes 16-31 -> M=r+8.
#pragma unroll
    for (int r = 0; r < 8; ++r)
        tile[(r + half * 8) * 16 + n] = acc[r];
    __syncthreads();

    if (lane < 16) {
        int m = tidx * 16 + lane;
        if (m < Q) {
            int best = 0;
            float bv = tile[lane * 16 + 0];
#pragma unroll
            for (int c = 1; c < NWAY; ++c) {
                float v = tile[lane * 16 + c];
                if (v > bv) { bv = v; best = c; }   // first-max tie rule
            }
            out[m] = (best == qy[m]) ? 1.0f : 0.0f;
        }
    }
}

// ---------------------------------------------------------------------------
extern "C" void kernel_launch(void* const* d_in, const int* in_sizes, int n_in,
                              void* d_out, int out_size, void* d_ws, size_t ws_size,
                              hipStream_t stream)
{
    const float* sup   = (const float*)d_in[0];   // support_xf (1,25,2048,5,5)
    const int*   sy    = (const int*)  d_in[1];   // support_y  (1,25)
    const float* qry   = (const float*)d_in[2];   // query_xf   (1,600,2048,5,5)
    const int*   qy    = (const int*)  d_in[3];   // query_y    (1,600)
    const int*   bids  = (const int*)  d_in[6];   // batch_ids  (700,4)
    const int*   bmask = (const int*)  d_in[7];   // batch_mask (700,4)
    const float* w0    = (const float*)d_in[8];   // weight_init (5,2048)
    const int n_steps  = in_sizes[6] / 4;

    float* ws = (float*)d_ws;
    float* qx = ws;                               // 600*2048
    float* sx = qx + (size_t)NQRY * CCH;          // 25*2048
    float* Wf = sx + (size_t)NSUP * CCH;          // 5*2048

    const int total = (NSUP + NQRY) * CCH;
    pool_kernel<<<(total + 255) / 256, 256, 0, stream>>>(sup, qry, sx, qx, total);
    train_kernel<<<1, 256, 0, stream>>>(sx, sy, bids, bmask, w0, Wf, n_steps);
    score_kernel<<<(NQRY + 15) / 16, 32, 0, stream>>>(qx, Wf, qy, (float*)d_out, NQRY);
}